// MarketGraphNet_45337674776979
// MI455X (gfx1250) — compile-verified
//
#include <hip/hip_runtime.h>
#include <math.h>

// ---------------------------------------------------------------------------
// Problem constants (from the reference)
// ---------------------------------------------------------------------------
#define NN      20000
#define EE      320000
#define IN_DIM  256
#define C1      512     // layer-1 hidden
#define C2      256     // layer-2 hidden
#define OUT_DIM 128
#define K1      10000   // ceil(N/2)
#define K2      5000    // ceil(K1/2)
#define EPSLN   1e-5f

typedef __attribute__((ext_vector_type(2))) float v2f;
typedef __attribute__((ext_vector_type(8))) float v8f;

// ---------------------------------------------------------------------------
// f32 WMMA GEMM:  D[M,Nc] = A[M,K] @ B[K,Nc]   (row-major, all f32)
// One wave computes a 16(M) x 64(N) strip = 4 x v_wmma_f32_16x16x4_f32 chains.
// Layouts per CDNA5 ISA 7.12.2:
//   A 16x4:  lanes 0-15 hold K={0,1}, lanes 16-31 hold K={2,3} (VGPR0/1)
//   B 4x16:  VGPR0 = rows K0(l0-15)/K2(l16-31), VGPR1 = rows K1/K3
//   C/D:     VGPR r -> M=r (lanes 0-15), M=r+8 (lanes 16-31)
// ---------------------------------------------------------------------------
__global__ __launch_bounds__(256) void gemm_wmma_f32(
    const float* __restrict__ A, const float* __restrict__ B,
    float* __restrict__ D, int M, int K, int Nc)
{
    const int lane   = threadIdx.x & 31;
    const int wave   = threadIdx.x >> 5;
    const int strips = Nc >> 6;                       // 64-wide N strips
    const int tiles  = ((M + 15) >> 4) * strips;
    const int tile   = blockIdx.x * 8 + wave;
    if (tile >= tiles) return;                        // wave-uniform exit

    const int mt   = tile / strips;
    const int ns   = tile % strips;
    const int m0   = mt * 16;
    const int n0   = ns * 64;
    const int half = lane >> 4;                       // 0: K0/1, 1: K2/3
    const int l15  = lane & 15;

    v8f acc0 = {}, acc1 = {}, acc2 = {}, acc3 = {};
    const float* arow = A + (long)(m0 + l15) * K;

    for (int k = 0; k < K; k += 4) {
        v2f a;
        a.x = arow[k + 2 * half + 0];
        a.y = arow[k + 2 * half + 1];
        const float* b0 = B + (long)(k + 2 * half + 0) * Nc + n0 + l15;
        const float* b1 = B + (long)(k + 2 * half + 1) * Nc + n0 + l15;
        v2f b;
        b.x = b0[0];  b.y = b1[0];
        acc0 = __builtin_amdgcn_wmma_f32_16x16x4_f32(false, a, false, b, (short)0, acc0, false, false);
        b.x = b0[16]; b.y = b1[16];
        acc1 = __builtin_amdgcn_wmma_f32_16x16x4_f32(false, a, false, b, (short)0, acc1, false, false);
        b.x = b0[32]; b.y = b1[32];
        acc2 = __builtin_amdgcn_wmma_f32_16x16x4_f32(false, a, false, b, (short)0, acc2, false, false);
        b.x = b0[48]; b.y = b1[48];
        acc3 = __builtin_amdgcn_wmma_f32_16x16x4_f32(false, a, false, b, (short)0, acc3, false, false);
    }

    float* dp = D + (long)(m0 + 8 * half) * Nc + n0 + l15;
    for (int r = 0; r < 8; ++r) {
        dp[0]  = acc0[r];
        dp[16] = acc1[r];
        dp[32] = acc2[r];
        dp[48] = acc3[r];
        dp += Nc;
    }
}

// ---------------------------------------------------------------------------
// Small utility kernels
// ---------------------------------------------------------------------------
__global__ void set_val_f(float* p, float v, int n) {
    int i = blockIdx.x * blockDim.x + threadIdx.x;
    if (i < n) p[i] = v;
}

__global__ void edge_deg(const int* __restrict__ dst, const float* __restrict__ w,
                         float* deg, int ne) {
    int e = blockIdx.x * blockDim.x + threadIdx.x;
    if (e >= ne) return;
    float we = w ? w[e] : 1.0f;
    if (we != 0.0f) atomicAdd(&deg[dst[e]], we);
}

__global__ void rsqrt_k(const float* deg, float* dinv, int n) {
    int i = blockIdx.x * blockDim.x + threadIdx.x;
    if (i < n) dinv[i] = rsqrtf(deg[i]);
}

// one wave per edge; scatter xW[src]*norm into out[dst]
__global__ __launch_bounds__(256) void edge_agg(
    const int* __restrict__ src, const int* __restrict__ dst,
    const float* __restrict__ w, const float* __restrict__ xW,
    float* __restrict__ out, const float* __restrict__ dinv, int ne, int C)
{
    int e = blockIdx.x * 8 + (threadIdx.x >> 5);
    if (e >= ne) return;
    float we = w ? w[e] : 1.0f;
    if (we == 0.0f) return;
    int s = src[e], d = dst[e];
    float nrm = dinv[s] * dinv[d] * we;
    int lane = threadIdx.x & 31;
    const float* xs = xW + (long)s * C;
    float* od = out + (long)d * C;
    for (int c = lane; c < C; c += 32) atomicAdd(&od[c], xs[c] * nrm);
}

// out += (2/deg)*xW + bias   (elementwise over n*C)
__global__ void finalize_conv(float* __restrict__ out, const float* __restrict__ xW,
                              const float* __restrict__ deg, const float* __restrict__ b,
                              long total, int C)
{
    long i = (long)blockIdx.x * blockDim.x + threadIdx.x;
    if (i >= total) return;
    int c = (int)(i % C);
    long node = i / C;
    out[i] += (2.0f / deg[node]) * xW[i] + b[c];
}

// grid-stride sum / sum-of-squares reduction -> pair[0], pair[1]
__global__ __launch_bounds__(256) void reduce_sum2(const float* __restrict__ x, long n,
                                                   float* pair)
{
    __shared__ float sh0[8], sh1[8];
    float s = 0.f, q = 0.f;
    for (long idx = (long)blockIdx.x * blockDim.x + threadIdx.x; idx < n;
         idx += (long)gridDim.x * blockDim.x) {
        float v = x[idx];
        s += v; q += v * v;
    }
    for (int o = 16; o > 0; o >>= 1) {
        s += __shfl_xor(s, o, 32);
        q += __shfl_xor(q, o, 32);
    }
    int lane = threadIdx.x & 31, wv = threadIdx.x >> 5;
    if (lane == 0) { sh0[wv] = s; sh1[wv] = q; }
    __syncthreads();
    if (threadIdx.x == 0) {
        float a = 0.f, b = 0.f;
        for (int w = 0; w < 8; ++w) { a += sh0[w]; b += sh1[w]; }
        atomicAdd(&pair[0], a);
        atomicAdd(&pair[1], b);
    }
}

// x = relu((x - mean)/(std+eps)*w + b), scalar mean/std over ALL entries
__global__ void norm_relu(float* __restrict__ x, const float* __restrict__ pair,
                          const float* __restrict__ w, const float* __restrict__ b,
                          long total, int C)
{
    long i = (long)blockIdx.x * blockDim.x + threadIdx.x;
    if (i >= total) return;
    float mean = pair[0] / (float)total;
    float var  = pair[1] / (float)total - mean * mean;
    float stdv = sqrtf(fmaxf(var, 0.0f));
    int c = (int)(i % C);
    float v = (x[i] - mean) / (stdv + EPSLN) * w[c] + b[c];
    x[i] = fmaxf(v, 0.0f);
}

// per-node dual dot products (wave per node):
//   srel[i] = h[i] . wrel ;  score[i] = h[i] . wroot + brel
__global__ __launch_bounds__(256) void node_scores(
    const float* __restrict__ h, const float* __restrict__ wrel,
    const float* __restrict__ wroot, const float* __restrict__ brel,
    float* srel, float* score, int n, int C)
{
    int node = blockIdx.x * 8 + (threadIdx.x >> 5);
    if (node >= n) return;
    int lane = threadIdx.x & 31;
    const float* hp = h + (long)node * C;
    float sr = 0.f, so = 0.f;
    for (int c = lane; c < C; c += 32) {
        float v = hp[c];
        sr += v * wrel[c];
        so += v * wroot[c];
    }
    for (int o = 16; o > 0; o >>= 1) {
        sr += __shfl_xor(sr, o, 32);
        so += __shfl_xor(so, o, 32);
    }
    if (lane == 0) { srel[node] = sr; score[node] = so + brel[0]; }
}

__global__ void edge_score(const int* __restrict__ src, const int* __restrict__ dst,
                           const float* __restrict__ w, const float* __restrict__ srel,
                           float* score, int ne)
{
    int e = blockIdx.x * blockDim.x + threadIdx.x;
    if (e >= ne) return;
    float we = w ? w[e] : 1.0f;
    if (we != 0.0f) atomicAdd(&score[dst[e]], srel[src[e]] * we);
}

__device__ __forceinline__ unsigned ordkey(float f) {
    unsigned u = __float_as_uint(f);
    return (u & 0x80000000u) ? ~u : (u | 0x80000000u);  // monotone: bigger f -> bigger key
}

__global__ void make_keys(const float* score, unsigned* key, int n) {
    int i = blockIdx.x * blockDim.x + threadIdx.x;
    if (i < n) key[i] = ordkey(score[i]);
}

__global__ void init_state(unsigned* state, unsigned k) {
    if (threadIdx.x == 0) { state[0] = 0u; state[1] = k; state[2] = 0u; }
}

// radix-select pass: histogram of byte at `shift` among keys matching prefix
__global__ void hist_pass(const unsigned* __restrict__ key, int n,
                          const unsigned* __restrict__ state, unsigned* hist, int shift)
{
    unsigned prefix = state[0];
    unsigned mask = (shift == 24) ? 0u : (0xFFFFFFFFu << (shift + 8));
    int i = blockIdx.x * blockDim.x + threadIdx.x;
    if (i < n) {
        unsigned k = key[i];
        if (((k ^ prefix) & mask) == 0u) atomicAdd(&hist[(k >> shift) & 255u], 1u);
    }
}

__global__ void radix_scan(const unsigned* hist, unsigned* state, int shift) {
    if (threadIdx.x != 0) return;
    unsigned kremain = state[1];
    unsigned acc = 0;
    for (int v = 255; v >= 0; --v) {
        unsigned h = hist[v];
        acc += h;
        if (acc >= kremain) {
            state[0] |= ((unsigned)v) << shift;
            state[1] = kremain - (acc - h);
            break;
        }
    }
}

// per-256-block counts of key>T and key==T
__global__ void blk_count(const unsigned* __restrict__ key, int n,
                          const unsigned* __restrict__ state,
                          unsigned* blkGt, unsigned* blkEq)
{
    __shared__ unsigned sg, se;
    if (threadIdx.x == 0) { sg = 0; se = 0; }
    __syncthreads();
    int i = blockIdx.x * 256 + threadIdx.x;
    unsigned T = state[0];
    if (i < n) {
        unsigned k = key[i];
        if (k > T) atomicAdd(&sg, 1u);
        else if (k == T) atomicAdd(&se, 1u);
    }
    __syncthreads();
    if (threadIdx.x == 0) { blkGt[blockIdx.x] = sg; blkEq[blockIdx.x] = se; }
}

__global__ void blk_scan(unsigned* blkGt, unsigned* blkEq, int nb, unsigned* state) {
    if (threadIdx.x != 0) return;
    unsigned a = 0, b = 0;
    for (int i = 0; i < nb; ++i) {
        unsigned g = blkGt[i], e = blkEq[i];
        blkGt[i] = a; blkEq[i] = b;
        a += g; b += e;
    }
    state[2] = a;   // total strictly-greater
}

// deterministic selection + dense relabeling:
//   key>T  -> newid = rank among gt by old index
//   key==T -> admitted while rankEq < kremain -> newid = totGt + rankEq
__global__ __launch_bounds__(256) void select_nodes(
    const unsigned* __restrict__ key, int n, const unsigned* __restrict__ state,
    const unsigned* __restrict__ offGt, const unsigned* __restrict__ offEq,
    int* kept, int* inv, int* oldOf)
{
    __shared__ unsigned wgt[8], weq[8];
    int i = blockIdx.x * 256 + threadIdx.x;
    unsigned T = state[0], krem = state[1], totGt = state[2];
    int lane = threadIdx.x & 31, wv = threadIdx.x >> 5;
    bool in = i < n;
    unsigned k = in ? key[i] : 0u;
    bool gt = in && (k > T);
    bool eq = in && (k == T);
    unsigned mgt = (unsigned)__ballot(gt);
    unsigned meq = (unsigned)__ballot(eq);
    if (lane == 0) { wgt[wv] = __popc(mgt); weq[wv] = __popc(meq); }
    __syncthreads();
    unsigned pg = 0, pe = 0;
    for (int w = 0; w < wv; ++w) { pg += wgt[w]; pe += weq[w]; }
    unsigned low = (1u << lane) - 1u;
    int newid = -1;
    if (gt) {
        newid = (int)(offGt[blockIdx.x] + pg + __popc(mgt & low));
    } else if (eq) {
        unsigned r = offEq[blockIdx.x] + pe + __popc(meq & low);
        if (r < krem) newid = (int)(totGt + r);
    }
    if (in) {
        kept[i] = (newid >= 0) ? 1 : 0;
        inv[i]  = (newid >= 0) ? newid : 0;
        if (newid >= 0) oldOf[newid] = i;
    }
}

// hp[new] = h[old] * tanh(score[old])   (wave per pooled node)
__global__ __launch_bounds__(256) void gather_pool(
    const int* __restrict__ oldOf, const float* __restrict__ h,
    const float* __restrict__ score, float* __restrict__ hp, int kk, int C)
{
    int j = blockIdx.x * 8 + (threadIdx.x >> 5);
    if (j >= kk) return;
    int i = oldOf[j];
    float t = tanhf(score[i]);
    int lane = threadIdx.x & 31;
    const float* s = h + (long)i * C;
    float* d = hp + (long)j * C;
    for (int c = lane; c < C; c += 32) d[c] = s[c] * t;
}

__global__ void remap_edges(const int* __restrict__ src, const int* __restrict__ dst,
                            const float* __restrict__ win,
                            const int* __restrict__ kept, const int* __restrict__ inv,
                            int* so, int* dо, float* wo, int ne)
{
    int e = blockIdx.x * blockDim.x + threadIdx.x;
    if (e >= ne) return;
    int s = src[e], d = dst[e];
    float we = win ? win[e] : 1.0f;
    float w = (we != 0.0f && kept[s] && kept[d]) ? we : 0.0f;
    so[e] = inv[s];
    dо[e] = inv[d];
    wo[e] = w;
}

// column mean over rows (single block, C threads)
__global__ void col_mean(const float* __restrict__ h, int rows, int C, float* g) {
    int c = threadIdx.x;
    if (c >= C) return;
    float s = 0.f;
    for (int r = 0; r < rows; ++r) s += h[(long)r * C + c];
    g[c] = s / (float)rows;
}

__global__ void final_linear(const float* __restrict__ g, const float* __restrict__ Wf,
                             const float* __restrict__ bf, float* out, int C, int Od)
{
    int j = blockIdx.x * blockDim.x + threadIdx.x;
    if (j >= Od) return;
    float s = bf[j];
    for (int c = 0; c < C; ++c) s += g[c] * Wf[c * Od + j];
    out[j] = s;
}

// ---------------------------------------------------------------------------
// Host-side stage helpers (all launches on stream; graph-capture safe)
// ---------------------------------------------------------------------------
static void run_conv(hipStream_t st, const float* h_in, int n, int Cin, int Cout,
                     const int* esrc, const int* edst, const float* ew, int ne,
                     const float* W, const float* bias, const float* lnw, const float* lnb,
                     float* xW, float* outbuf, float* deg, float* dinv, float* pair)
{
    long total = (long)n * Cout;
    hipMemsetAsync(outbuf, 0, total * sizeof(float), st);
    set_val_f<<<(n + 255) / 256, 256, 0, st>>>(deg, 2.0f, n);
    edge_deg<<<(ne + 255) / 256, 256, 0, st>>>(edst, ew, deg, ne);
    {   // GEMM: xW = h_in @ W
        int strips = Cout >> 6;
        int tiles  = ((n + 15) >> 4) * strips;
        gemm_wmma_f32<<<(tiles + 7) / 8, 256, 0, st>>>(h_in, W, xW, n, Cin, Cout);
    }
    rsqrt_k<<<(n + 255) / 256, 256, 0, st>>>(deg, dinv, n);
    edge_agg<<<(ne + 7) / 8, 256, 0, st>>>(esrc, edst, ew, xW, outbuf, dinv, ne, Cout);
    finalize_conv<<<(int)((total + 255) / 256), 256, 0, st>>>(outbuf, xW, deg, bias, total, Cout);
    hipMemsetAsync(pair, 0, 2 * sizeof(float), st);
    {
        long nb = (total + 255) / 256;
        if (nb > 4096) nb = 4096;
        reduce_sum2<<<(int)nb, 256, 0, st>>>(outbuf, total, pair);
    }
    norm_relu<<<(int)((total + 255) / 256), 256, 0, st>>>(outbuf, pair, lnw, lnb, total, Cout);
}

static void run_pool(hipStream_t st, const float* h, int n, int C, int k,
                     const float* wrel, const float* brel, const float* wroot,
                     const int* esrc, const int* edst, const float* ew, int ne,
                     float* srel, float* score, unsigned* key, unsigned* hist,
                     unsigned* state, unsigned* blkGt, unsigned* blkEq,
                     int* kept, int* inv, int* oldOf, float* hpool)
{
    node_scores<<<(n + 7) / 8, 256, 0, st>>>(h, wrel, wroot, brel, srel, score, n, C);
    edge_score<<<(ne + 255) / 256, 256, 0, st>>>(esrc, edst, ew, srel, score, ne);
    make_keys<<<(n + 255) / 256, 256, 0, st>>>(score, key, n);
    init_state<<<1, 1, 0, st>>>(state, (unsigned)k);
    for (int shift = 24; shift >= 0; shift -= 8) {
        hipMemsetAsync(hist, 0, 256 * sizeof(unsigned), st);
        hist_pass<<<(n + 255) / 256, 256, 0, st>>>(key, n, state, hist, shift);
        radix_scan<<<1, 1, 0, st>>>(hist, state, shift);
    }
    int nb = (n + 255) / 256;
    blk_count<<<nb, 256, 0, st>>>(key, n, state, blkGt, blkEq);
    blk_scan<<<1, 1, 0, st>>>(blkGt, blkEq, nb, state);
    select_nodes<<<nb, 256, 0, st>>>(key, n, state, blkGt, blkEq, kept, inv, oldOf);
    gather_pool<<<(k + 7) / 8, 256, 0, st>>>(oldOf, h, score, hpool, k, C);
}

// ---------------------------------------------------------------------------
extern "C" void kernel_launch(void* const* d_in, const int* in_sizes, int n_in,
                              void* d_out, int out_size, void* d_ws, size_t ws_size,
                              hipStream_t stream)
{
    (void)in_sizes; (void)n_in; (void)out_size; (void)ws_size;

    const float* x       = (const float*)d_in[0];
    const int*   ei      = (const int*)d_in[1];
    const float* W1      = (const float*)d_in[2];
    const float* b1      = (const float*)d_in[3];
    const float* ln1w    = (const float*)d_in[4];
    const float* ln1b    = (const float*)d_in[5];
    const float* p1wrel  = (const float*)d_in[6];
    const float* p1brel  = (const float*)d_in[7];
    const float* p1wroot = (const float*)d_in[8];
    const float* W2      = (const float*)d_in[9];
    const float* b2      = (const float*)d_in[10];
    const float* ln2w    = (const float*)d_in[11];
    const float* ln2b    = (const float*)d_in[12];
    const float* p2wrel  = (const float*)d_in[13];
    const float* p2brel  = (const float*)d_in[14];
    const float* p2wroot = (const float*)d_in[15];
    const float* Wf      = (const float*)d_in[16];
    const float* bf      = (const float*)d_in[17];
    float* out = (float*)d_out;

    const int* src = ei;
    const int* dst = ei + EE;

    // ---- bump allocator over d_ws ----
    char* wsp = (char*)d_ws;
    auto alloc = [&](size_t bytes) -> void* {
        void* p = (void*)wsp;
        wsp += (bytes + 255) & ~(size_t)255;
        return p;
    };
    float*    xW1   = (float*)alloc((size_t)NN * C1 * 4);
    float*    out1  = (float*)alloc((size_t)NN * C1 * 4);
    float*    h1p   = (float*)alloc((size_t)K1 * C1 * 4);
    float*    xW2   = (float*)alloc((size_t)K1 * C2 * 4);
    float*    out2  = (float*)alloc((size_t)K1 * C2 * 4);
    float*    h2p   = (float*)alloc((size_t)K2 * C2 * 4);
    float*    deg1  = (float*)alloc((size_t)NN * 4);
    float*    dinv1 = (float*)alloc((size_t)NN * 4);
    float*    deg2  = (float*)alloc((size_t)K1 * 4);
    float*    dinv2 = (float*)alloc((size_t)K1 * 4);
    float*    srel  = (float*)alloc((size_t)NN * 4);
    float*    score = (float*)alloc((size_t)NN * 4);
    unsigned* key   = (unsigned*)alloc((size_t)NN * 4);
    int*      kept  = (int*)alloc((size_t)NN * 4);
    int*      inv   = (int*)alloc((size_t)NN * 4);
    int*      oldOf = (int*)alloc((size_t)K1 * 4);
    int*      src2  = (int*)alloc((size_t)EE * 4);
    int*      dst2  = (int*)alloc((size_t)EE * 4);
    float*    w2    = (float*)alloc((size_t)EE * 4);
    unsigned* hist  = (unsigned*)alloc(256 * 4);
    unsigned* state = (unsigned*)alloc(4 * 4);
    unsigned* blkGt = (unsigned*)alloc(128 * 4);
    unsigned* blkEq = (unsigned*)alloc(128 * 4);
    float*    pair  = (float*)alloc(2 * 4);
    float*    g     = (float*)alloc(C2 * 4);

    // ===== Layer 1: GCNConv(256->512) + GraphNorm + ReLU =====
    run_conv(stream, x, NN, IN_DIM, C1, src, dst, /*w=*/nullptr, EE,
             W1, b1, ln1w, ln1b, xW1, out1, deg1, dinv1, pair);

    // ===== SAGPool 1: N -> K1, relabel edges =====
    run_pool(stream, out1, NN, C1, K1, p1wrel, p1brel, p1wroot,
             src, dst, nullptr, EE,
             srel, score, key, hist, state, blkGt, blkEq, kept, inv, oldOf, h1p);
    remap_edges<<<(EE + 255) / 256, 256, 0, stream>>>(src, dst, nullptr, kept, inv,
                                                      src2, dst2, w2, EE);

    // ===== Layer 2: GCNConv(512->256) + GraphNorm + ReLU =====
    run_conv(stream, h1p, K1, C1, C2, src2, dst2, w2, EE,
             W2, b2, ln2w, ln2b, xW2, out2, deg2, dinv2, pair);

    // ===== SAGPool 2: K1 -> K2 (edges not needed afterwards) =====
    run_pool(stream, out2, K1, C2, K2, p2wrel, p2brel, p2wroot,
             src2, dst2, w2, EE,
             srel, score, key, hist, state, blkGt, blkEq, kept, inv, oldOf, h2p);

    // ===== global mean pool + final linear =====
    col_mean<<<1, C2, 0, stream>>>(h2p, K2, C2, g);
    final_linear<<<1, OUT_DIM, 0, stream>>>(g, Wf, bf, out, C2, OUT_DIM);
}